// GCN2_4956392259903
// MI455X (gfx1250) — compile-verified
//
#include <hip/hip_runtime.h>
#include <stdint.h>

#define IN_DIM 128
#define HID    96
#define OUTD   64

typedef __attribute__((ext_vector_type(2))) float v2f;
typedef __attribute__((ext_vector_type(8))) float v8f;

// ---------------------------------------------------------------- utilities

__global__ void gcn_zero_kernel(float* __restrict__ p, int n) {
    int i = blockIdx.x * blockDim.x + threadIdx.x;
    if (i < n) p[i] = 0.0f;
}

__global__ void gcn_deg_kernel(const long long* __restrict__ dst,
                               float* __restrict__ deg, int E) {
    int e = blockIdx.x * blockDim.x + threadIdx.x;
    if (e < E) atomicAdd(deg + (int)dst[e], 1.0f);
}

__global__ void gcn_dinv_kernel(float* __restrict__ deg, int n) {
    int i = blockIdx.x * blockDim.x + threadIdx.x;
    if (i < n) {
        float d = deg[i];
        deg[i] = (d > 0.0f) ? rsqrtf(d) : 0.0f;   // deg>0 -> rsqrt(max(deg,1)) == rsqrt(deg)
    }
}

__global__ void gcn_norm_kernel(const long long* __restrict__ src,
                                const long long* __restrict__ dst,
                                const float* __restrict__ dinv,
                                float* __restrict__ norm, int E) {
    int e = blockIdx.x * blockDim.x + threadIdx.x;
    if (e < E) norm[e] = dinv[(int)src[e]] * dinv[(int)dst[e]];
}

// ------------------------------------------------- WMMA fp32 GEMM: C = A @ B
// A: [M,K] row-major, B: [K,NC] row-major, C: [M,NC] row-major.
// One wave32 per 16x16 C tile, K-loop of V_WMMA_F32_16X16X4_F32.
template <int K, int NC>
__global__ void __launch_bounds__(256)
gcn_gemm_wmma_f32(const float* __restrict__ A, const float* __restrict__ B,
                  float* __restrict__ C, int M) {
    const int lane = threadIdx.x & 31;
    const int wave = (blockIdx.x * blockDim.x + threadIdx.x) >> 5;
    constexpr int NT = NC / 16;
    const int mt = wave / NT;
    const int nt = wave - mt * NT;
    if (mt * 16 >= M) return;
    const int m0 = mt * 16;
    const int n0 = nt * 16;

    const int r  = lane & 15;          // row (A) / col (B,C) within tile
    const int kb = (lane >> 4) << 1;   // lanes 0-15 -> K {0,1}; lanes 16-31 -> K {2,3}

    const float* Arow = A + (size_t)(m0 + r) * K;   // A[m0+r][*]
    const float* Bcol = B + n0 + r;                 // B[*][n0+r]

    v8f acc = {};
#pragma unroll
    for (int k = 0; k < K; k += 4) {
        v2f a, b;
        a.x = Arow[k + kb];
        a.y = Arow[k + kb + 1];
        b.x = Bcol[(size_t)(k + kb) * NC];
        b.y = Bcol[(size_t)(k + kb + 1) * NC];
        // (neg_a, A, neg_b, B, c_mod, C, reuse_a, reuse_b)
        acc = __builtin_amdgcn_wmma_f32_16x16x4_f32(false, a, false, b,
                                                    (short)0, acc, false, false);
    }

    // C/D layout: VGPR j -> M = j + 8*(lane>=16), N = lane&15
    const int mb = m0 + ((lane >> 4) << 3);
    float* Cp = C + (size_t)mb * NC + n0 + (lane & 15);
#pragma unroll
    for (int j = 0; j < 8; ++j) Cp[(size_t)j * NC] = acc[j];
}

// --------------------------------------- edge scatter: agg[dst] += h[src]*w
// One thread per (edge, 4-float chunk); float4 load + 4 global f32 atomics.
template <int F>
__global__ void gcn_scatter_kernel(const long long* __restrict__ src,
                                   const long long* __restrict__ dst,
                                   const float* __restrict__ norm,
                                   const float* __restrict__ h,
                                   float* __restrict__ agg, int E) {
    constexpr int CH = F / 4;
    int tid = blockIdx.x * blockDim.x + threadIdx.x;
    if (tid >= E * CH) return;
    int e = tid / CH;
    int c = tid - e * CH;
    int s = (int)src[e];
    int d = (int)dst[e];
    float w = norm[e];
    const float4 v = ((const float4*)(h + (size_t)s * F))[c];
    float* o = agg + (size_t)d * F + c * 4;
    atomicAdd(o + 0, v.x * w);
    atomicAdd(o + 1, v.y * w);
    atomicAdd(o + 2, v.z * w);
    atomicAdd(o + 3, v.w * w);
}

__global__ void gcn_bias_relu_kernel(float* __restrict__ h,
                                     const float* __restrict__ b, int n, int F) {
    int i = blockIdx.x * blockDim.x + threadIdx.x;
    if (i < n) {
        float v = h[i] + b[i % F];
        h[i] = v > 0.0f ? v : 0.0f;
    }
}

__global__ void gcn_bias_kernel(float* __restrict__ h,
                                const float* __restrict__ b, int n, int F) {
    int i = blockIdx.x * blockDim.x + threadIdx.x;
    if (i < n) h[i] += b[i % F];
}

// ------------------------------------------------------------------- launch

static inline int nblk(long long n, int t) { return (int)((n + t - 1) / t); }

extern "C" void kernel_launch(void* const* d_in, const int* in_sizes, int n_in,
                              void* d_out, int out_size, void* d_ws, size_t ws_size,
                              hipStream_t stream) {
    const float*     x   = (const float*)d_in[0];
    const long long* ei  = (const long long*)d_in[1];   // int64 [2,E]
    const float*     W1  = (const float*)d_in[2];
    const float*     b1  = (const float*)d_in[3];
    const float*     W2  = (const float*)d_in[4];
    const float*     b2  = (const float*)d_in[5];

    const int N = in_sizes[0] / IN_DIM;
    const int E = in_sizes[1] / 2;
    const long long* src = ei;
    const long long* dst = ei + E;
    float* out = (float*)d_out;

    // workspace layout (floats): dinv[N] | norm[E] | hpre[N*HID] | agg1[N*HID]
    float* ws   = (float*)d_ws;
    float* dinv = ws;
    float* norm = dinv + N;
    float* hpre = norm + E;                    // reused as layer-2 pre [N*OUTD]
    float* agg1 = hpre + (size_t)N * HID;      // becomes h1 after bias+relu

    const int T = 256;

    // --- normalization coefficients ---
    gcn_zero_kernel<<<nblk(N, T), T, 0, stream>>>(dinv, N);
    gcn_deg_kernel<<<nblk(E, T), T, 0, stream>>>(dst, dinv, E);
    gcn_dinv_kernel<<<nblk(N, T), T, 0, stream>>>(dinv, N);
    gcn_norm_kernel<<<nblk(E, T), T, 0, stream>>>(src, dst, dinv, norm, E);

    // --- layer 1: hpre = x @ W1 ; agg1 = scatter(hpre*norm) ; h1 = relu(agg1+b1)
    {
        long long waves = (long long)((N + 15) / 16) * (HID / 16);
        gcn_gemm_wmma_f32<IN_DIM, HID>
            <<<nblk(waves * 32, T), T, 0, stream>>>(x, W1, hpre, N);
    }
    gcn_zero_kernel<<<nblk((long long)N * HID, T), T, 0, stream>>>(agg1, N * HID);
    gcn_scatter_kernel<HID>
        <<<nblk((long long)E * (HID / 4), T), T, 0, stream>>>(src, dst, norm, hpre, agg1, E);
    gcn_bias_relu_kernel<<<nblk((long long)N * HID, T), T, 0, stream>>>(agg1, b1, N * HID, HID);

    // --- layer 2: hpre = h1 @ W2 ; out = scatter(hpre*norm) + b2
    {
        long long waves = (long long)((N + 15) / 16) * (OUTD / 16);
        gcn_gemm_wmma_f32<HID, OUTD>
            <<<nblk(waves * 32, T), T, 0, stream>>>(agg1, W2, hpre, N);
    }
    gcn_zero_kernel<<<nblk((long long)N * OUTD, T), T, 0, stream>>>(out, N * OUTD);
    gcn_scatter_kernel<OUTD>
        <<<nblk((long long)E * (OUTD / 4), T), T, 0, stream>>>(src, dst, norm, hpre, out, E);
    gcn_bias_kernel<<<nblk((long long)N * OUTD, T), T, 0, stream>>>(out, b2, N * OUTD, OUTD);
}